// Qwen3_5MoE_33028298506814
// MI455X (gfx1250) — compile-verified
//
#include <hip/hip_runtime.h>
#include <math.h>

// MoE: B=128, H=2048, E=64, I=768, TOPK=8, fp32 throughout.
// Strategy: route -> expert-grouped fp32 WMMA GEMMs (w1+silu*up fused, then w2),
// deterministic per-(token,k) scatter + fixed-order reduce.
// Grid order: m-tile is blockIdx.x so all m-tiles sharing one weight tile are
// dispatched adjacently -> weight re-reads hit L2 (192 MB), HBM traffic ~= the
// 1.2 GB compulsory weight stream (~52 us @ 23.3 TB/s).

#define B_TOK 128
#define HDIM  2048
#define NEXP  64
#define IDIM  768
#define TOPK  8
#define CAP   128   // max tokens per expert

typedef float v2f __attribute__((ext_vector_type(2)));
typedef float v8f __attribute__((ext_vector_type(8)));

// ---------------- Kernel 1: gating / routing ----------------
// grid: (B_TOK) blocks of 64 threads. Thread e computes logit for expert e.
__global__ void moe_gate_kernel(const float* __restrict__ x,
                                const float* __restrict__ gate_w,
                                int* __restrict__ cnt,
                                int* __restrict__ slot_token,
                                int* __restrict__ slot_k,
                                float* __restrict__ slot_w) {
    __shared__ float logits[NEXP];
    const int b = blockIdx.x;
    const int e = threadIdx.x;

    const float4* xr = (const float4*)(x + (size_t)b * HDIM);
    const float4* gr = (const float4*)(gate_w + (size_t)e * HDIM);
    float acc = 0.f;
    #pragma unroll 4
    for (int k = 0; k < HDIM / 4; ++k) {
        float4 xv = xr[k], gv = gr[k];
        acc += xv.x * gv.x + xv.y * gv.y + xv.z * gv.z + xv.w * gv.w;
    }
    logits[e] = acc;
    __syncthreads();

    if (threadIdx.x == 0) {
        // top-8 by logits (same order as softmax probs); renormalized weights:
        // softmax denominator cancels -> w_k = exp(l_k - max)/sum over top8.
        int idx[TOPK]; float val[TOPK];
        unsigned long long used = 0ull;
        for (int k = 0; k < TOPK; ++k) {
            float best = -INFINITY; int bi = 0;
            for (int j = 0; j < NEXP; ++j) {
                if (!((used >> j) & 1ull) && logits[j] > best) { best = logits[j]; bi = j; }
            }
            used |= (1ull << bi); idx[k] = bi; val[k] = best;
        }
        const float m = val[0];
        float w[TOPK]; float s = 0.f;
        for (int k = 0; k < TOPK; ++k) { w[k] = expf(val[k] - m); s += w[k]; }
        const float inv = 1.f / s;
        for (int k = 0; k < TOPK; ++k) {
            const int e2 = idx[k];
            const int slot = atomicAdd(&cnt[e2], 1);
            slot_token[e2 * CAP + slot] = b;
            slot_k    [e2 * CAP + slot] = k;
            slot_w    [e2 * CAP + slot] = w[k] * inv;
        }
    }
}

// ---------------- Kernel 2: W1 GEMM + SiLU*up ----------------
// grid: (CAP/16, IDIM/16, NEXP), block: 32 (one wave).
// D(16x16) = A(16 tokens x K) * B(K x 16 intermediate cols), K stepped by 4.
// A layout (f32 16x4): lanes 0-15 hold K=0,1 ; lanes 16-31 hold K=2,3 (row M = lane&15).
// B layout (f32 4x16): VGPR j: lanes 0-15 -> K=j, N=lane ; lanes 16-31 -> K=j+2, N=lane-16.
__global__ void moe_ffn1_kernel(const float* __restrict__ x,
                                const float* __restrict__ w1,
                                const int* __restrict__ cnt,
                                const int* __restrict__ slot_token,
                                float* __restrict__ inter) {
    const int mt = blockIdx.x;          // m tile (fastest: weight-tile L2 reuse)
    const int it = blockIdx.y;          // intermediate tile: i0 = it*16
    const int e  = blockIdx.z;
    const int n  = cnt[e];
    const int m0 = mt * 16;
    if (m0 >= n) return;

    const int lane = threadIdx.x;
    const int half = lane >> 4;         // 0: K=0,1 ; 1: K=2,3
    const int l15  = lane & 15;

    const int arow = m0 + l15;
    const int tok  = (arow < n) ? slot_token[e * CAP + arow] : 0;
    const float* ap = x + (size_t)tok * HDIM + 2 * half;

    const size_t w1e = (size_t)e * (2 * IDIM) * HDIM;
    const float* bg = w1 + w1e + (size_t)(it * 16 + l15) * HDIM + 2 * half;        // gate rows
    const float* bu = bg + (size_t)IDIM * HDIM;                                     // up rows

    v8f cg = {};
    v8f cu = {};
    #pragma unroll 4
    for (int k0 = 0; k0 < HDIM; k0 += 4) {
        v2f a = *(const v2f*)(ap + k0);
        v2f g = *(const v2f*)(bg + k0);
        v2f u = *(const v2f*)(bu + k0);
        cg = __builtin_amdgcn_wmma_f32_16x16x4_f32(false, a, false, g, (short)0, cg, false, false);
        cu = __builtin_amdgcn_wmma_f32_16x16x4_f32(false, a, false, u, (short)0, cu, false, false);
    }

    // C/D layout: VGPR v -> row (v + 8*half), col = l15. inter[e][row][i0+col].
    #pragma unroll
    for (int v = 0; v < 8; ++v) {
        const int r = m0 + v + 8 * half;
        if (r < n) {
            const float g = cg[v];
            const float silu = g / (1.f + expf(-g));
            inter[((size_t)e * CAP + r) * IDIM + it * 16 + l15] = silu * cu[v];
        }
    }
}

// ---------------- Kernel 3: W2 GEMM, scale by routing weight ----------------
// grid: (CAP/16, HDIM/16, NEXP), block: 32.
__global__ void moe_ffn2_kernel(const float* __restrict__ inter,
                                const float* __restrict__ w2,
                                const int* __restrict__ cnt,
                                const int* __restrict__ slot_token,
                                const int* __restrict__ slot_k,
                                const float* __restrict__ slot_w,
                                float* __restrict__ outk) {
    const int mt = blockIdx.x;          // m tile (fastest: weight-tile L2 reuse)
    const int ht = blockIdx.y;          // h0 = ht*16
    const int e  = blockIdx.z;
    const int n  = cnt[e];
    const int m0 = mt * 16;
    if (m0 >= n) return;

    const int lane = threadIdx.x;
    const int half = lane >> 4;
    const int l15  = lane & 15;

    const float* ap = inter + ((size_t)e * CAP + m0 + l15) * IDIM + 2 * half;
    const float* bp = w2 + (size_t)e * HDIM * IDIM + (size_t)(ht * 16 + l15) * IDIM + 2 * half;

    v8f c0 = {};
    v8f c1 = {};
    #pragma unroll 4
    for (int k0 = 0; k0 < IDIM; k0 += 8) {
        v2f a0 = *(const v2f*)(ap + k0);
        v2f b0 = *(const v2f*)(bp + k0);
        v2f a1 = *(const v2f*)(ap + k0 + 4);
        v2f b1 = *(const v2f*)(bp + k0 + 4);
        c0 = __builtin_amdgcn_wmma_f32_16x16x4_f32(false, a0, false, b0, (short)0, c0, false, false);
        c1 = __builtin_amdgcn_wmma_f32_16x16x4_f32(false, a1, false, b1, (short)0, c1, false, false);
    }

    #pragma unroll
    for (int v = 0; v < 8; ++v) {
        const int r = m0 + v + 8 * half;
        if (r < n) {
            const int   tok = slot_token[e * CAP + r];
            const int   kk  = slot_k    [e * CAP + r];
            const float w   = slot_w    [e * CAP + r];
            outk[((size_t)tok * TOPK + kk) * HDIM + ht * 16 + l15] = w * (c0[v] + c1[v]);
        }
    }
}

// ---------------- Kernel 4: deterministic 8-way reduce ----------------
__global__ void moe_reduce_kernel(const float* __restrict__ outk, float* __restrict__ out) {
    const int i = blockIdx.x * blockDim.x + threadIdx.x;
    if (i >= B_TOK * HDIM) return;
    const int b = i >> 11;       // /2048
    const int h = i & (HDIM - 1);
    float s = 0.f;
    #pragma unroll
    for (int k = 0; k < TOPK; ++k) s += outk[((size_t)b * TOPK + k) * HDIM + h];
    out[i] = s;
}

extern "C" void kernel_launch(void* const* d_in, const int* in_sizes, int n_in,
                              void* d_out, int out_size, void* d_ws, size_t ws_size,
                              hipStream_t stream) {
    const float* x      = (const float*)d_in[0];   // (128, 2048)
    const float* gate_w = (const float*)d_in[1];   // (64, 2048)
    const float* w1     = (const float*)d_in[2];   // (64, 1536, 2048)
    const float* w2     = (const float*)d_in[3];   // (64, 2048, 768)
    float* out = (float*)d_out;                    // (128, 1, 2048)

    // Workspace layout (256B aligned chunks)
    char* ws = (char*)d_ws;
    int*   cnt        = (int*)(ws + 0);                          // 64 ints
    int*   slot_token = (int*)(ws + 256);                        // 64*128 ints
    int*   slot_k     = (int*)(ws + 256 + 32768);                // 64*128 ints
    float* slot_w     = (float*)(ws + 256 + 2 * 32768);          // 64*128 floats
    float* inter      = (float*)(ws + 256 + 3 * 32768);          // 64*128*768 floats (25.2 MB)
    float* outk       = (float*)(ws + 256 + 3 * 32768 + (size_t)NEXP * CAP * IDIM * 4); // 128*8*2048 (8.4 MB)

    hipMemsetAsync(cnt, 0, NEXP * sizeof(int), stream);

    moe_gate_kernel<<<B_TOK, NEXP, 0, stream>>>(x, gate_w, cnt, slot_token, slot_k, slot_w);

    moe_ffn1_kernel<<<dim3(CAP / 16, IDIM / 16, NEXP), 32, 0, stream>>>(
        x, w1, cnt, slot_token, inter);

    moe_ffn2_kernel<<<dim3(CAP / 16, HDIM / 16, NEXP), 32, 0, stream>>>(
        inter, w2, cnt, slot_token, slot_k, slot_w, outk);

    moe_reduce_kernel<<<(B_TOK * HDIM + 255) / 256, 256, 0, stream>>>(outk, out);
}